// SparseGraphAttention_15590731285216
// MI455X (gfx1250) — compile-verified
//
#include <hip/hip_runtime.h>
#include <hip/hip_bf16.h>
#include <math.h>

// ---------------- problem constants (match reference) ----------------
constexpr int N     = 50000;   // total nodes (multiple of 16: 3125 tiles)
constexpr int E     = 800000;  // edges
constexpr int HID   = 128;     // hidden
constexpr int HEADS = 8;
constexpr int DH    = HID / HEADS;       // 16
constexpr float SCALE = 0.25f;           // 1/sqrt(16)

typedef float v2f __attribute__((ext_vector_type(2)));
typedef float v8f __attribute__((ext_vector_type(8)));

// ---------------- helpers ----------------
// Exact float atomic-max via monotonic integer bit-pattern mapping:
// for v >= 0 the int pattern orders like the float; for v < 0 the unsigned
// pattern orders inversely. Both lower to single native integer atomics
// (no CAS loop). Compatible with -INFINITY (0xFF800000) initialization.
__device__ __forceinline__ void atomMaxF(float* p, float v) {
  if (v >= 0.0f) atomicMax((int*)p, __float_as_int(v));
  else           atomicMin((unsigned int*)p, __float_as_uint(v));
}
__device__ __forceinline__ void atomAddF(float* p, float v) {
  unsafeAtomicAdd(p, v);   // native global_atomic_add_f32 on CDNA
}

// ---------------- init: mx = -inf, se = 0 ----------------
__global__ void init_mx_se(float* __restrict__ mx, float* __restrict__ se) {
  int i = blockIdx.x * blockDim.x + threadIdx.x;
  if (i < N * HEADS) { mx[i] = -INFINITY; se[i] = 0.0f; }
}

__global__ void zero_buf(float* __restrict__ p, long long n) {
  long long i = (long long)blockIdx.x * blockDim.x + threadIdx.x;
  if (i < n) p[i] = 0.0f;
}

// ---------------- WMMA f32 GEMM core ----------------
// Per-wave: one 16x16 output tile of Y = X @ W^T + b, K looped 128/4 = 32
// v_wmma_f32_16x16x4_f32. LDS pitch 132 floats: ds bank = (k + 4*idx) % 64,
// conflict-free for both A and B fragment reads.
constexpr int LDP = 132;

__device__ __forceinline__ void wmma_tile_compute(
    const float* __restrict__ Xs, const float* __restrict__ Ws,
    const float* __restrict__ bias, float* __restrict__ Y,
    int row0, int nrows, int tid) {
  const int lane = tid & 31;
  const int wave = tid >> 5;          // 0..7 -> col tile
  const int l    = lane & 15;
  const int kh   = (lane >> 4) << 1;  // 0 for lanes 0-15, 2 for lanes 16-31
  const int colb = wave * 16;

  v8f acc = {0.f, 0.f, 0.f, 0.f, 0.f, 0.f, 0.f, 0.f};
  #pragma unroll
  for (int k0 = 0; k0 < HID; k0 += 4) {
    v2f a, b;
    // A 16x4 (MxK): lane l holds row l; VGPR0=K(kh), VGPR1=K(kh+1)
    a.x = Xs[l * LDP + k0 + kh];
    a.y = Xs[l * LDP + k0 + kh + 1];
    // B 4x16 (KxN): B[k][n] = W[n][k]; lane l holds col colb+l
    b.x = Ws[(colb + l) * LDP + k0 + kh];
    b.y = Ws[(colb + l) * LDP + k0 + kh + 1];
    acc = __builtin_amdgcn_wmma_f32_16x16x4_f32(false, a, false, b,
                                                (short)0, acc, false, false);
  }
  // C/D layout: VGPR r -> (M=r, N=lane) lanes 0-15, (M=r+8, N=lane-16) lanes 16-31
  const int mbase = (lane < 16) ? 0 : 8;
  const int col   = colb + l;
  const float bv  = bias[col];
  #pragma unroll
  for (int r = 0; r < 8; ++r) {
    int row = row0 + mbase + r;
    if (row < nrows) Y[(long long)row * HID + col] = acc[r] + bv;
  }
}

// Fused Q/K/V projection: stage the 16x128 X tile once, loop over 3 weights.
__global__ void __launch_bounds__(256)
qkv_gemm_wmma(const float* __restrict__ X,
              const float* __restrict__ Wq, const float* __restrict__ bq, float* __restrict__ Q,
              const float* __restrict__ Wk, const float* __restrict__ bk, float* __restrict__ K,
              const float* __restrict__ Wv, const float* __restrict__ bv, float* __restrict__ V,
              int nrows) {
  __shared__ float Xs[16 * LDP];    //  ~8.4 KB
  __shared__ float Ws[128 * LDP];   // ~67.6 KB
  const int tid  = threadIdx.x;
  const int row0 = blockIdx.x * 16;

  #pragma unroll
  for (int i = tid; i < 16 * HID; i += 256) {
    int r = i >> 7, k = i & 127;
    Xs[r * LDP + k] = X[(long long)(row0 + r) * HID + k];
  }

  const float* Wp[3] = {Wq, Wk, Wv};
  const float* bp[3] = {bq, bk, bv};
  float*       Yp[3] = {Q,  K,  V};

  #pragma unroll
  for (int j = 0; j < 3; ++j) {
    __syncthreads();   // Xs ready (j=0) / previous compute done before Ws reuse
    #pragma unroll
    for (int i = tid; i < HID * HID; i += 256) {
      int r = i >> 7, k = i & 127;
      Ws[r * LDP + k] = Wp[j][i];
    }
    __syncthreads();
    wmma_tile_compute(Xs, Ws, bp[j], Yp[j], row0, nrows, tid);
  }
}

// Single GEMM (output projection).
__global__ void __launch_bounds__(256)
gemm_wmma_f32(const float* __restrict__ X, const float* __restrict__ W,
              const float* __restrict__ bias, float* __restrict__ Y, int nrows) {
  __shared__ float Xs[16 * LDP];
  __shared__ float Ws[128 * LDP];
  const int tid  = threadIdx.x;
  const int row0 = blockIdx.x * 16;

  #pragma unroll
  for (int i = tid; i < 16 * HID; i += 256) {
    int r = i >> 7, k = i & 127;
    Xs[r * LDP + k] = X[(long long)(row0 + r) * HID + k];
  }
  #pragma unroll
  for (int i = tid; i < HID * HID; i += 256) {
    int r = i >> 7, k = i & 127;
    Ws[r * LDP + k] = W[i];
  }
  __syncthreads();
  wmma_tile_compute(Xs, Ws, bias, Y, row0, nrows, tid);
}

// ---------------- edge scores + segment max ----------------
// one thread per (edge, head); 8 consecutive threads share an edge -> each
// reads a contiguous 64B slice of a Q/K row (coalesced within the row).
__global__ void edge_scores(const float* __restrict__ Q, const float* __restrict__ K,
                            const long long* __restrict__ ei,
                            float* __restrict__ S, float* __restrict__ mx) {
  long long t = (long long)blockIdx.x * blockDim.x + threadIdx.x;
  if (t >= (long long)E * HEADS) return;
  int e = (int)(t >> 3), h = (int)(t & 7);
  int src = (int)ei[e];
  int dst = (int)ei[E + e];
  const float4* q4 = (const float4*)(Q + (long long)dst * HID + h * DH);
  const float4* k4 = (const float4*)(K + (long long)src * HID + h * DH);
  float s = 0.f;
  #pragma unroll
  for (int i = 0; i < 4; ++i) {
    float4 a = q4[i], b = k4[i];
    s += a.x * b.x + a.y * b.y + a.z * b.z + a.w * b.w;
  }
  s *= SCALE;
  S[t] = s;
  atomMaxF(&mx[dst * HEADS + h], s);
}

// ---------------- exp + segment sum ----------------
__global__ void edge_exp(const long long* __restrict__ ei,
                         float* __restrict__ S, const float* __restrict__ mx,
                         float* __restrict__ se) {
  long long t = (long long)blockIdx.x * blockDim.x + threadIdx.x;
  if (t >= (long long)E * HEADS) return;
  int e = (int)(t >> 3), h = (int)(t & 7);
  int dst = (int)ei[E + e];
  float ex = __expf(S[t] - mx[dst * HEADS + h]);
  S[t] = ex;
  atomAddF(&se[dst * HEADS + h], ex);
}

// ---------------- weighted scatter-add of V ----------------
// one thread per (edge, h, d): 128 consecutive threads cover a full V row
// (fully coalesced); agg scatter via native f32 atomics (L2-resident).
__global__ void edge_aggregate(const long long* __restrict__ ei,
                               const float* __restrict__ S,
                               const float* __restrict__ se,
                               const float* __restrict__ V,
                               float* __restrict__ agg) {
  long long t = (long long)blockIdx.x * blockDim.x + threadIdx.x;
  if (t >= (long long)E * HID) return;
  int e = (int)(t >> 7);
  int c = (int)(t & 127);          // c = h*16 + d
  int h = c >> 4;
  int src = (int)ei[e];
  int dst = (int)ei[E + e];
  float attn = S[(long long)e * HEADS + h] / (se[dst * HEADS + h] + 1e-12f);
  atomAddF(&agg[(long long)dst * HID + c], attn * V[(long long)src * HID + c]);
}

// ---------------- launch ----------------
extern "C" void kernel_launch(void* const* d_in, const int* in_sizes, int n_in,
                              void* d_out, int out_size, void* d_ws, size_t ws_size,
                              hipStream_t stream) {
  const float*     x  = (const float*)d_in[0];
  const long long* ei = (const long long*)d_in[1];   // int64 edge_index (2 x E)
  const float* Wq = (const float*)d_in[3];
  const float* bq = (const float*)d_in[4];
  const float* Wk = (const float*)d_in[5];
  const float* bk = (const float*)d_in[6];
  const float* Wv = (const float*)d_in[7];
  const float* bv = (const float*)d_in[8];
  const float* Wo = (const float*)d_in[9];
  const float* bo = (const float*)d_in[10];
  float* out = (float*)d_out;

  // workspace layout (floats): Q | K | V | S(E*H) | mx | se ; agg reuses K
  float* ws = (float*)d_ws;
  const long long NH = (long long)N * HID;
  float* Q   = ws;
  float* Kb  = Q + NH;
  float* V   = Kb + NH;
  float* S   = V + NH;
  float* mx  = S + (long long)E * HEADS;
  float* se  = mx + (long long)N * HEADS;
  float* agg = Kb;   // K dead after edge_scores

  const int TB = 256;
  const int rowTiles = N / 16;                                         // 3125
  const int ehBlocks = (int)(((long long)E * HEADS + TB - 1) / TB);    // 25000
  const int edBlocks = (int)(((long long)E * HID   + TB - 1) / TB);    // 400000

  init_mx_se<<<(N * HEADS + TB - 1) / TB, TB, 0, stream>>>(mx, se);

  qkv_gemm_wmma<<<rowTiles, TB, 0, stream>>>(x, Wq, bq, Q, Wk, bk, Kb,
                                             Wv, bv, V, N);

  edge_scores<<<ehBlocks, TB, 0, stream>>>(Q, Kb, ei, S, mx);
  edge_exp   <<<ehBlocks, TB, 0, stream>>>(ei, S, mx, se);

  zero_buf<<<(int)((NH + TB - 1) / TB), TB, 0, stream>>>(agg, NH);
  edge_aggregate<<<edBlocks, TB, 0, stream>>>(ei, S, se, V, agg);

  gemm_wmma_f32<<<rowTiles, TB, 0, stream>>>(agg, Wo, bo, out, N);
}